// MSeE_49117245997815
// MI455X (gfx1250) — compile-verified
//
#include <hip/hip_runtime.h>
#include <hip/hip_bf16.h>

// ---------------------------------------------------------------------------
// filtfilt(order-2 IIR) + MSE for [256, 65536] f32, tuned for MI455X (gfx1250)
//  * one workgroup per row, entire row resident in 320KB WGP LDS (256KB row)
//  * IIR parallelized as chunked scan: 128 chunks x 512, exact stitch via
//    homogeneous-response tables (poles 0.997/0.32 -> stitching mandatory)
//  * chunk-local recurrence on the matrix pipe: V_WMMA_F32_16X16X4_F32
//    blocked Toeplitz formulation, 5 WMMAs / 16 timesteps / 16 chunks (f32!)
//  * backward pass = reversed logical indexing (no physical flip)
//  * filtered-predictions scratch (64MB) lives in the 192MB L2
// ---------------------------------------------------------------------------

typedef float v2f __attribute__((ext_vector_type(2)));
typedef float v8f __attribute__((ext_vector_type(8)));

#if defined(__gfx1250__) && __has_builtin(__builtin_amdgcn_wmma_f32_16x16x4_f32)
#define USE_WMMA_F32 1
#else
#define USE_WMMA_F32 0
#endif

#define T_LEN    65536
#define B_ROWS   256
#define CHUNK    512
#define NCHUNK   128          // T_LEN / CHUNK
#define NTHREADS 256          // 8 wave32 -> 8*16 = 128 WMMA columns = chunks

// lfilter coefficients (a0 == 1)
#define A1c (-1.31861375911f)
#define A2c ( 0.32059452332f)
#define B0c ( 0.95616638497f)
#define B1c (-1.31960414122f)
#define B2c ( 0.36343775625f)

// d_ws layout (float offsets)
#define WS_P1    0            // [512]  homogeneous response to y[-1]=1
#define WS_P2    512          // [512]  homogeneous response to y[-2]=1
#define WS_T     1024         // [256]  16x16 lower-tri Toeplitz impulse matrix
#define WS_PART  2048         // [256]  per-row partial sums
#define WS_FILTP 4096         // [256*65536] filtered predictions (64MB, fits L2)

// LDS layout (float offsets), total 67328 floats = 263.0 KB <= 320 KB/WGP
#define L_BUF   0
#define L_P1    65536
#define L_P2    66048
#define L_E2    66560         // [2*NCHUNK] particular end states (z_end, z_end-1)
#define L_SENT  66816         // [2*NCHUNK] true entering states per chunk
#define L_RED   67072         // [NTHREADS] reduction scratch
#define LDS_FLOATS 67328

__device__ __forceinline__ float clamp1(float v) {
    return fminf(fmaxf(v, -1.0f), 1.0f);
}

// ---------------------------------------------------------------------------
// Kernel 0: build constant tables (impulse-response Toeplitz + homogeneous
// solutions). Tiny serial work, identical every launch (deterministic).
// ---------------------------------------------------------------------------
__global__ void mse_build_tables(float* __restrict__ ws) {
    if (threadIdx.x != 0 || blockIdx.x != 0) return;
    float h[16];
    h[0] = 1.0f;
    h[1] = -A1c;
    for (int n = 2; n < 16; ++n) h[n] = -A1c * h[n - 1] - A2c * h[n - 2];
    for (int i = 0; i < 16; ++i)
        for (int j = 0; j < 16; ++j)
            ws[WS_T + i * 16 + j] = (i >= j) ? h[i - j] : 0.0f;
    {   // P1[n]: response with y[-1]=1, y[-2]=0, f=0
        float y1 = 1.0f, y2 = 0.0f;
        for (int n = 0; n < CHUNK; ++n) {
            float y = -A1c * y1 - A2c * y2;
            ws[WS_P1 + n] = y; y2 = y1; y1 = y;
        }
    }
    {   // P2[n]: response with y[-1]=0, y[-2]=1
        float y1 = 0.0f, y2 = 1.0f;
        for (int n = 0; n < CHUNK; ++n) {
            float y = -A1c * y1 - A2c * y2;
            ws[WS_P2 + n] = y; y2 = y1; y1 = y;
        }
    }
}

// ---------------------------------------------------------------------------
// Chunked IIR scan, in place on LDS buf (FIR values in, filtered values out).
// REV=true runs the recurrence over logical index l with phys = T-1-l
// (backward filtfilt pass) while keeping all LDS accesses aligned b64/b128.
// ---------------------------------------------------------------------------
template <bool REV>
__device__ __forceinline__ void iir_scan(float* buf, const float* p1c, const float* p2c,
                                         float* e2, float* sEnt,
                                         const v2f* aT, v2f aU) {
    const int tid = threadIdx.x;
    __syncthreads();

#if USE_WMMA_F32
    // ---- matrix-pipe chunk scan: 16 chunks per wave as the 16 B-columns ----
    const int  lane  = tid & 31;
    const int  lhalf = lane & 15;
    const bool hi    = lane >= 16;
    const int  col   = (tid >> 5) * 16 + lhalf;   // chunk owned by this column

    v2f bcar = {0.0f, 0.0f};                      // carry rows (y[15], y[14])
    for (int bs = 0; bs < CHUNK; bs += 16) {
        // B tiles: FIR values f[bs+4j .. bs+4j+3] of each chunk (column)
        v2f bt[4];
#pragma unroll
        for (int j = 0; j < 4; ++j) {
            int l0 = col * CHUNK + bs + 4 * j + (hi ? 2 : 0);  // logical pair
            if (!REV) {
                bt[j] = *(const v2f*)(buf + l0);
            } else {
                v2f t = *(const v2f*)(buf + (T_LEN - 2 - l0));
                bt[j].x = t.y; bt[j].y = t.x;                  // reverse pair
            }
        }
        v8f d = {0.f, 0.f, 0.f, 0.f, 0.f, 0.f, 0.f, 0.f};
        // carry: U(16x2, cols g1,g2; K2/K3 zero) x S  -- junk in hi-lane B rows
        // is multiplied by U's zero columns, so the shfl garbage is harmless.
        d = __builtin_amdgcn_wmma_f32_16x16x4_f32(false, aU,    false, bcar,  (short)0, d, false, false);
        // Toeplitz: y_block += T(16x16) * f_block, as 4 K=4 tiles
        d = __builtin_amdgcn_wmma_f32_16x16x4_f32(false, aT[0], false, bt[0], (short)0, d, false, false);
        d = __builtin_amdgcn_wmma_f32_16x16x4_f32(false, aT[1], false, bt[1], (short)0, d, false, false);
        d = __builtin_amdgcn_wmma_f32_16x16x4_f32(false, aT[2], false, bt[2], (short)0, d, false, false);
        d = __builtin_amdgcn_wmma_f32_16x16x4_f32(false, aT[3], false, bt[3], (short)0, d, false, false);

        // store particular solution over the FIR values (8 consecutive / lane)
        int lb = col * CHUNK + bs + (hi ? 8 : 0);              // logical base
        if (!REV) {
            *(float4*)(buf + lb)     = make_float4(d[0], d[1], d[2], d[3]);
            *(float4*)(buf + lb + 4) = make_float4(d[4], d[5], d[6], d[7]);
        } else {
            *(float4*)(buf + (T_LEN - 4 - lb)) = make_float4(d[3], d[2], d[1], d[0]);
            *(float4*)(buf + (T_LEN - 8 - lb)) = make_float4(d[7], d[6], d[5], d[4]);
        }
        // carry (y[15], y[14]) of each column from hi half-wave -> lanes 0-15
        bcar.x = __shfl_xor(d[7], 16, 32);
        bcar.y = __shfl_xor(d[6], 16, 32);
        if (bs == CHUNK - 16 && hi) {          // chunk end particular state
            e2[2 * col]     = d[7];            // z[511]
            e2[2 * col + 1] = d[6];            // z[510]
        }
    }
#else
    // ---- VALU fallback: one thread serially scans one chunk ----
    (void)aT; (void)aU;
    if (tid < NCHUNK) {
        const int c = tid;
        float y1 = 0.0f, y2v = 0.0f;
        for (int n = 0; n < CHUNK; ++n) {
            int   l  = c * CHUNK + n;
            int   ph = REV ? (T_LEN - 1 - l) : l;
            float z  = buf[ph] - A1c * y1 - A2c * y2v;
            buf[ph]  = z;
            y2v = y1; y1 = z;
        }
        e2[2 * c] = y1; e2[2 * c + 1] = y2v;
    }
#endif
    __syncthreads();

    // ---- exact boundary stitch: 128-step 2x2 recursion on one thread ----
    if (tid == 0) {
        const float g11 = p1c[CHUNK - 1], g12 = p2c[CHUNK - 1];
        const float g21 = p1c[CHUNK - 2], g22 = p2c[CHUNK - 2];
        float sl = 0.0f, sp = 0.0f;                 // entering (y[-1], y[-2])
        for (int c = 0; c < NCHUNK; ++c) {
            sEnt[2 * c] = sl; sEnt[2 * c + 1] = sp;
            float nl = e2[2 * c]     + g11 * sl + g12 * sp;   // true y[end]
            float np = e2[2 * c + 1] + g21 * sl + g22 * sp;   // true y[end-1]
            sl = nl; sp = np;
        }
    }
    __syncthreads();

    // ---- vectorized correction: y = z + s1*P1[ofs] + s2*P2[ofs] ----
    for (int base = tid * 4; base < T_LEN; base += NTHREADS * 4) {
        int   c   = base >> 9;
        int   ofs = base & (CHUNK - 1);
        float sl  = sEnt[2 * c], sp = sEnt[2 * c + 1];
        if (!REV) {
            float4 z = *(float4*)(buf + base);
            z.x += sl * p1c[ofs]     + sp * p2c[ofs];
            z.y += sl * p1c[ofs + 1] + sp * p2c[ofs + 1];
            z.z += sl * p1c[ofs + 2] + sp * p2c[ofs + 2];
            z.w += sl * p1c[ofs + 3] + sp * p2c[ofs + 3];
            *(float4*)(buf + base) = z;
        } else {
            float4 z = *(float4*)(buf + (T_LEN - 4 - base)); // .w == logical base
            z.w += sl * p1c[ofs]     + sp * p2c[ofs];
            z.z += sl * p1c[ofs + 1] + sp * p2c[ofs + 1];
            z.y += sl * p1c[ofs + 2] + sp * p2c[ofs + 2];
            z.x += sl * p1c[ofs + 3] + sp * p2c[ofs + 3];
            *(float4*)(buf + (T_LEN - 4 - base)) = z;
        }
    }
    __syncthreads();
}

// ---------------------------------------------------------------------------
// Full filtfilt of one row, in LDS. Output lands in buf, natural orientation,
// unclamped (clamp applied at consumption, matching the reference).
// ---------------------------------------------------------------------------
__device__ void filtfilt_row(const float* __restrict__ xr, float* buf,
                             const float* p1c, const float* p2c,
                             float* e2, float* sEnt, const v2f* aT, v2f aU) {
    const int tid = threadIdx.x;

    // ---- forward FIR while staging from global (coalesced float4) ----
    for (int it = 0; it < T_LEN / (NTHREADS * 16); ++it) {
        int seg = (it * NTHREADS + tid) * 16;
        if (it + 1 < T_LEN / (NTHREADS * 16))
            __builtin_prefetch(xr + seg + NTHREADS * 16, 0, 1); // global_prefetch_b8
        float xm2 = (seg >= 2) ? xr[seg - 2] : 0.0f;
        float xm1 = (seg >= 1) ? xr[seg - 1] : 0.0f;
        float xv[16];
#pragma unroll
        for (int b = 0; b < 4; ++b) {
            float4 q = *(const float4*)(xr + seg + 4 * b);
            xv[4 * b] = q.x; xv[4 * b + 1] = q.y; xv[4 * b + 2] = q.z; xv[4 * b + 3] = q.w;
        }
        float f[16];
        float pm1 = xm1, pm2 = xm2;
#pragma unroll
        for (int k = 0; k < 16; ++k) {
            f[k] = B0c * xv[k] + B1c * pm1 + B2c * pm2;
            pm2 = pm1; pm1 = xv[k];
        }
#pragma unroll
        for (int b = 0; b < 4; ++b)
            *(float4*)(buf + seg + 4 * b) =
                make_float4(f[4 * b], f[4 * b + 1], f[4 * b + 2], f[4 * b + 3]);
    }

    iir_scan<false>(buf, p1c, p2c, e2, sEnt, aT, aU);

    // ---- backward FIR in natural orientation (local +2 halo, in place) ----
    // f2_phys[n] = b0*y[n] + b1*y[n+1] + b2*y[n+2]  (zero past the end)
    for (int it = 0; it < T_LEN / (NTHREADS * 16); ++it) {
        int seg = (it * NTHREADS + tid) * 16;
        float v[18];
#pragma unroll
        for (int k = 0; k < 16; ++k) v[k] = buf[seg + k];
        v[16] = (seg + 16 < T_LEN) ? buf[seg + 16] : 0.0f;
        v[17] = (seg + 17 < T_LEN) ? buf[seg + 17] : 0.0f;
        __syncthreads();                       // all reads before any writes
#pragma unroll
        for (int k = 0; k < 16; ++k)
            buf[seg + k] = B0c * v[k] + B1c * v[k + 1] + B2c * v[k + 2];
        __syncthreads();
    }

    iir_scan<true>(buf, p1c, p2c, e2, sEnt, aT, aU);
    // buf now holds filtfilt(x) in natural orientation (flip-back implicit).
}

// ---------------------------------------------------------------------------
// Kernel 1: one workgroup per row; filter predictions (-> L2 scratch) and
// targets (LDS), accumulate per-row sum of squared clamped differences.
// ---------------------------------------------------------------------------
__global__ void __launch_bounds__(NTHREADS)
mse_filtfilt_rows(const float* __restrict__ pred, const float* __restrict__ targ,
                  float* __restrict__ ws) {
    extern __shared__ float lds[];
    float* buf  = lds + L_BUF;
    float* p1c  = lds + L_P1;
    float* p2c  = lds + L_P2;
    float* e2   = lds + L_E2;
    float* sEnt = lds + L_SENT;
    float* red  = lds + L_RED;

    const int tid = threadIdx.x;
    const int r   = blockIdx.x;

    for (int i = tid; i < CHUNK; i += NTHREADS) {
        p1c[i] = ws[WS_P1 + i];
        p2c[i] = ws[WS_P2 + i];
    }

    // constant WMMA A-tiles: Toeplitz column slices + carry matrix U
    v2f aT[4] = {{0.f, 0.f}, {0.f, 0.f}, {0.f, 0.f}, {0.f, 0.f}};
    v2f aU    = {0.f, 0.f};
#if USE_WMMA_F32
    {
        const int  lane  = tid & 31;
        const int  lhalf = lane & 15;
        const bool hi    = lane >= 16;
        const float* Tt  = ws + WS_T;
#pragma unroll
        for (int j = 0; j < 4; ++j) {
            int o = lhalf * 16 + 4 * j + (hi ? 2 : 0);
            aT[j].x = Tt[o]; aT[j].y = Tt[o + 1];
        }
        if (!hi) { aU.x = ws[WS_P1 + lhalf]; aU.y = ws[WS_P2 + lhalf]; }
    }
#endif
    __syncthreads();

    float* fp = ws + WS_FILTP + (size_t)r * T_LEN;

    // ---- predictions: filtfilt -> clamp -> L2-resident scratch ----
    filtfilt_row(pred + (size_t)r * T_LEN, buf, p1c, p2c, e2, sEnt, aT, aU);
    for (int i = tid * 4; i < T_LEN; i += NTHREADS * 4) {
        float4 v = *(float4*)(buf + i);
        v.x = clamp1(v.x); v.y = clamp1(v.y); v.z = clamp1(v.z); v.w = clamp1(v.w);
        *(float4*)(fp + i) = v;
    }
    __syncthreads();

    // ---- targets: filtfilt in LDS, accumulate (p - t)^2 ----
    filtfilt_row(targ + (size_t)r * T_LEN, buf, p1c, p2c, e2, sEnt, aT, aU);
    float acc = 0.0f;
    for (int i = tid * 4; i < T_LEN; i += NTHREADS * 4) {
        float4 pv = *(const float4*)(fp + i);     // same-thread store->load order
        float4 tv = *(float4*)(buf + i);
        float dx = pv.x - clamp1(tv.x);
        float dy = pv.y - clamp1(tv.y);
        float dz = pv.z - clamp1(tv.z);
        float dw = pv.w - clamp1(tv.w);
        acc += dx * dx + dy * dy + dz * dz + dw * dw;
    }
    red[tid] = acc;
    __syncthreads();
    for (int s = NTHREADS / 2; s > 0; s >>= 1) {
        if (tid < s) red[tid] += red[tid + s];
        __syncthreads();
    }
    if (tid == 0) ws[WS_PART + r] = red[0];
}

// ---------------------------------------------------------------------------
// Kernel 2: reduce per-row partials to the scalar mean.
// ---------------------------------------------------------------------------
__global__ void mse_finalize(const float* __restrict__ ws, float* __restrict__ out) {
    __shared__ float red[NTHREADS];
    const int tid = threadIdx.x;
    red[tid] = ws[WS_PART + tid];
    __syncthreads();
    for (int s = NTHREADS / 2; s > 0; s >>= 1) {
        if (tid < s) red[tid] += red[tid + s];
        __syncthreads();
    }
    if (tid == 0)
        out[0] = red[0] * (1.0f / ((float)B_ROWS * (float)T_LEN));
}

extern "C" void kernel_launch(void* const* d_in, const int* in_sizes, int n_in,
                              void* d_out, int out_size, void* d_ws, size_t ws_size,
                              hipStream_t stream) {
    (void)in_sizes; (void)n_in; (void)out_size; (void)ws_size;
    const float* pred = (const float*)d_in[0];
    const float* targ = (const float*)d_in[1];
    float*       ws   = (float*)d_ws;   // needs (WS_FILTP + 256*65536)*4 ~ 64MB+16KB

    mse_build_tables<<<1, 1, 0, stream>>>(ws);

    const size_t ldsBytes = (size_t)LDS_FLOATS * sizeof(float);  // 263 KB / WGP
    mse_filtfilt_rows<<<B_ROWS, NTHREADS, ldsBytes, stream>>>(pred, targ, ws);

    mse_finalize<<<1, NTHREADS, 0, stream>>>(ws, (float*)d_out);
}